// GCN_2688649527830
// MI455X (gfx1250) — compile-verified
//
#include <hip/hip_runtime.h>
#include <hip/hip_bf16.h>

// ---------------------------------------------------------------------------
// GCN 2-layer forward on MI455X (gfx1250).
//   h1 = relu( A_norm @ (X @ W1) + b1 )        A_norm = D^-1/2 (A+I) D^-1/2
//   h2 =        A_norm @ (h1 @ W2) + b2
//   out = h2 @ Wfc + bfc
// GEMMs use V_WMMA_F32_16X16X4_F32 (full fp32 precision, matches reference).
// Sparse aggregation uses global_atomic_add_f32 (memory-bound part, ~2.6 GB
// of HBM traffic per call -> the roofline cost; WMMA GEMMs are negligible).
// ---------------------------------------------------------------------------

typedef float v2f __attribute__((ext_vector_type(2)));
typedef float v8f __attribute__((ext_vector_type(8)));

#define IN_C  128
#define HID_C 128
#define OUT_C 64
#define NCLS  2

// ---------------------------------------------------------------- degree ops
__global__ void k_deg_init(float* __restrict__ deg, int n) {
    int i = blockIdx.x * blockDim.x + threadIdx.x;
    if (i < n) deg[i] = 1.0f;            // self-loop contributes 1
}

__global__ void k_deg_count(const int* __restrict__ dst, float* __restrict__ deg, int e) {
    int i = blockIdx.x * blockDim.x + threadIdx.x;
    if (i < e) atomicAdd(&deg[dst[i]], 1.0f);
}

__global__ void k_rsqrt(float* __restrict__ d, int n) {
    int i = blockIdx.x * blockDim.x + threadIdx.x;
    if (i < n) d[i] = rsqrtf(d[i]);
}

// ------------------------------------------------------------- WMMA fp32 GEMM
// H[nrows x OUTC] = X[nrows x 128] * W[128 x OUTC]
// block = 32 * (OUTC/16) threads; one block handles 16 rows, each wave one
// 16-column tile. X tile staged in LDS (padded stride 132 -> conflict-free
// reads of the 16x4 fp32 A-fragment layout).
template <int OUTC>
__global__ void __launch_bounds__(32 * (OUTC / 16))
k_gemm_wmma(const float* __restrict__ X, const float* __restrict__ W,
            float* __restrict__ H) {
    constexpr int KW   = 128;
    constexpr int LDSW = 132;               // padded row stride (floats)
    constexpr int NT   = 32 * (OUTC / 16);
    __shared__ float lds[16 * LDSW];

    const int row0 = blockIdx.x * 16;
    const int tid  = threadIdx.x;

    // Stage 16x128 fp32 tile of X into LDS, coalesced float4 loads.
    for (int i = tid; i < 16 * (KW / 4); i += NT) {
        int r  = i >> 5;                    // / (KW/4)
        int c4 = i & 31;
        float4 v = ((const float4*)(X + (size_t)(row0 + r) * KW))[c4];
        float* d = &lds[r * LDSW + c4 * 4];
        d[0] = v.x; d[1] = v.y; d[2] = v.z; d[3] = v.w;
    }
    __syncthreads();

    const int wave  = tid >> 5;
    const int lane  = tid & 31;
    const int nlane = lane & 15;            // N (col) within tile / M (row) for A
    const int khalf = (lane >> 4) * 2;      // lanes 16-31 handle K+2, K+3

    v8f acc = {};                           // 16x16 fp32 accumulator tile

    for (int k0 = 0; k0 < KW; k0 += 4) {
        // A fragment (16x4 fp32): lane<16 -> A[m=lane][k0+0..1],
        //                         lane>=16 -> A[m=lane-16][k0+2..3]
        const float* ap = &lds[nlane * LDSW + k0 + khalf];
        v2f a; a.x = ap[0]; a.y = ap[1];
        // B fragment (4x16 fp32): lane<16 -> B[k0+0..1][n],
        //                         lane>=16 -> B[k0+2..3][n]
        const float* bp = W + (size_t)(k0 + khalf) * OUTC + wave * 16 + nlane;
        v2f b; b.x = bp[0]; b.y = bp[OUTC];
        acc = __builtin_amdgcn_wmma_f32_16x16x4_f32(
            /*neg_a=*/false, a, /*neg_b=*/false, b,
            /*c_mod=*/(short)0, acc, /*reuse_a=*/false, /*reuse_b=*/false);
    }

    // D layout: VGPR r -> row r (lanes 0-15) / row r+8 (lanes 16-31), col nlane.
    const int rbase = (lane >> 4) * 8;
    float* out = H + (size_t)(row0 + rbase) * OUTC + wave * 16 + nlane;
#pragma unroll
    for (int r = 0; r < 8; ++r) out[(size_t)r * OUTC] = acc[r];
}

// ----------------------------------------------------- sparse aggregation ops
// Self-loop term, full overwrite (doubles as the zero-init of agg):
// agg[i][c] = dinv[i]^2 * h[i][c]
template <int C>
__global__ void k_selfloop(const float* __restrict__ h, const float* __restrict__ dinv,
                           float* __restrict__ agg, int n) {
    int idx = blockIdx.x * blockDim.x + threadIdx.x;
    if (idx >= n * C) return;
    int i = idx / C;
    float di = dinv[i];
    agg[idx] = di * di * h[idx];
}

// Edge scatter: agg[dst] += dinv[src]*dinv[dst] * h[src], float4 per thread.
template <int C>
__global__ void k_edge_scatter(const int* __restrict__ src, const int* __restrict__ dst,
                               const float* __restrict__ dinv,
                               const float* __restrict__ h, float* __restrict__ agg,
                               int e) {
    constexpr int G = C / 4;                // float4 groups per edge
    unsigned int t = blockIdx.x * blockDim.x + threadIdx.x;
    if (t >= (unsigned int)e * G) return;
    int ed = t / G;
    int g  = t % G;
    int s = src[ed], d = dst[ed];
    float nrm = dinv[s] * dinv[d];
    float4 hv = ((const float4*)(h + (size_t)s * C))[g];
    float* o = agg + (size_t)d * C + g * 4;
    atomicAdd(o + 0, hv.x * nrm);
    atomicAdd(o + 1, hv.y * nrm);
    atomicAdd(o + 2, hv.z * nrm);
    atomicAdd(o + 3, hv.w * nrm);
}

// h = relu(agg + b), in place over agg.
template <int C>
__global__ void k_bias_relu(float* __restrict__ agg, const float* __restrict__ b, int n) {
    int idx = blockIdx.x * blockDim.x + threadIdx.x;
    if (idx >= n * C) return;
    agg[idx] = fmaxf(agg[idx] + b[idx % C], 0.0f);
}

// out[i] = (agg2[i] + b2) @ Wfc + bfc      (64 -> 2)
__global__ void k_final_fc(const float* __restrict__ agg2, const float* __restrict__ b2,
                           const float* __restrict__ Wfc, const float* __restrict__ bfc,
                           float* __restrict__ out, int n) {
    int i = blockIdx.x * blockDim.x + threadIdx.x;
    if (i >= n) return;
    const float* row = agg2 + (size_t)i * OUT_C;
    float a0 = bfc[0], a1 = bfc[1];
#pragma unroll 8
    for (int c = 0; c < OUT_C; ++c) {
        float v = row[c] + b2[c];
        a0 = fmaf(v, Wfc[c * 2 + 0], a0);
        a1 = fmaf(v, Wfc[c * 2 + 1], a1);
    }
    out[i * 2 + 0] = a0;
    out[i * 2 + 1] = a1;
}

// ---------------------------------------------------------------------------
extern "C" void kernel_launch(void* const* d_in, const int* in_sizes, int n_in,
                              void* d_out, int out_size, void* d_ws, size_t ws_size,
                              hipStream_t stream) {
    const float* x    = (const float*)d_in[0];
    const int*   ei   = (const int*)  d_in[1];   // [2, E] int32
    const float* W1   = (const float*)d_in[2];
    const float* b1   = (const float*)d_in[3];
    const float* W2   = (const float*)d_in[4];
    const float* b2   = (const float*)d_in[5];
    const float* Wfc  = (const float*)d_in[6];
    const float* bfc  = (const float*)d_in[7];
    float*       out  = (float*)d_out;

    const int N = in_sizes[0] / IN_C;            // 100000
    const int E = in_sizes[1] / 2;               // 1600000
    const int* src = ei;
    const int* dst = ei + E;

    // Workspace layout (floats): dinv[N] | H[N*128] | AGG[N*128] | AGG2[N*64]
    float* dinv = (float*)d_ws;
    float* H    = dinv + N;
    float* AGG  = H + (size_t)N * HID_C;
    float* AGG2 = AGG + (size_t)N * HID_C;

    const int BT = 256;

    // --- degrees / normalization ---
    k_deg_init<<<(N + BT - 1) / BT, BT, 0, stream>>>(dinv, N);
    k_deg_count<<<(E + BT - 1) / BT, BT, 0, stream>>>(dst, dinv, E);
    k_rsqrt<<<(N + BT - 1) / BT, BT, 0, stream>>>(dinv, N);

    // --- layer 1: H = X @ W1 (WMMA fp32), aggregate, bias+relu ---
    k_gemm_wmma<HID_C><<<N / 16, 32 * (HID_C / 16), 0, stream>>>(x, W1, H);
    {
        int tot = N * HID_C;
        k_selfloop<HID_C><<<(tot + BT - 1) / BT, BT, 0, stream>>>(H, dinv, AGG, N);
        unsigned int work = (unsigned int)E * (HID_C / 4);
        k_edge_scatter<HID_C><<<(work + BT - 1) / BT, BT, 0, stream>>>(src, dst, dinv, H, AGG, E);
        k_bias_relu<HID_C><<<(tot + BT - 1) / BT, BT, 0, stream>>>(AGG, b1, N);
    }

    // --- layer 2: H2 = h1 @ W2 (WMMA fp32), aggregate ---
    k_gemm_wmma<OUT_C><<<N / 16, 32 * (OUT_C / 16), 0, stream>>>(AGG, W2, H);
    {
        int tot = N * OUT_C;
        k_selfloop<OUT_C><<<(tot + BT - 1) / BT, BT, 0, stream>>>(H, dinv, AGG2, N);
        unsigned int work = (unsigned int)E * (OUT_C / 4);
        k_edge_scatter<OUT_C><<<(work + BT - 1) / BT, BT, 0, stream>>>(src, dst, dinv, H, AGG2, E);
    }

    // --- final FC (bias2 fused) ---
    k_final_fc<<<(N + BT - 1) / BT, BT, 0, stream>>>(AGG2, b2, Wfc, bfc, out, N);
}